// KGAT_15281493639475
// MI455X (gfx1250) — compile-verified
//
#include <hip/hip_runtime.h>
#include <hip/hip_bf16.h>
#include <math.h>
#include <stdint.h>

// ---------------------------------------------------------------------------
// CDNA5 (gfx1250) KGAT pipeline.  Dense matrix math: V_WMMA_F32_16X16X4_F32.
// Tile staging: GLOBAL_LOAD_ASYNC_TO_LDS_B128 (ASYNCcnt) + s_wait_asynccnt.
// ---------------------------------------------------------------------------

typedef __attribute__((ext_vector_type(2))) float v2f;
typedef __attribute__((ext_vector_type(8))) float v8f;

#define ENTITY_OFFSET 20000
#define EMB_D 64
#define CAT_D 160

__device__ __forceinline__ v8f wmma_f32_16x16x4(v2f a, v2f b, v8f c) {
    // 8 args: (neg_a, A, neg_b, B, c_mod, C, reuse_a, reuse_b)
    return __builtin_amdgcn_wmma_f32_16x16x4_f32(false, a, false, b,
                                                 (short)0, c, false, false);
}

__device__ __forceinline__ float wave_sum32(float x) {
    #pragma unroll
    for (int o = 16; o > 0; o >>= 1) x += __shfl_xor(x, o, 32);
    return x;
}

__device__ __forceinline__ float leaky01(float x) {
    return x > 0.0f ? x : 0.01f * x;
}

// Async 16B global->LDS copy (CDNA5 ASYNCcnt path, bypasses VGPRs).
// lds_off = LDS byte address (low 32 bits of the generic shared address),
// gsrc    = 16B-aligned global source.
__device__ __forceinline__ void async_copy_b128(unsigned lds_off,
                                                const float* gsrc) {
    asm volatile("global_load_async_to_lds_b128 %0, %1, off"
                 :: "v"(lds_off), "v"(gsrc) : "memory");
}

__device__ __forceinline__ void wait_asynccnt0() {
    asm volatile("s_wait_asynccnt 0x0" ::: "memory");
}

// ---------------------------------------------------------------------------
// Kernel 1: gate = sigmoid(log1p(aux) @ W_aux.T + b_aux)*0.15 + 1
//           ego  = l2norm(entity * gate) + type_embed[node >= OFFSET]
// One wave per node; each lane owns dims {lane, lane+32}.
// ---------------------------------------------------------------------------
__global__ __launch_bounds__(256)
void k_fuse(const float* __restrict__ aux, const float* __restrict__ entity,
            const float* __restrict__ type_embed, const float* __restrict__ W_aux,
            const float* __restrict__ b_aux, float* __restrict__ ego,
            float* __restrict__ all_e, int N) {
    int gt = blockIdx.x * blockDim.x + threadIdx.x;
    int node = gt >> 5;
    if (node >= N) return;
    int lane = threadIdx.x & 31;

    float a0 = log1pf(aux[node * 3 + 0]);
    float a1 = log1pf(aux[node * 3 + 1]);
    float a2 = log1pf(aux[node * 3 + 2]);

    float e[2];
    #pragma unroll
    for (int h = 0; h < 2; ++h) {
        int d = lane + 32 * h;
        float z = b_aux[d] + a0 * W_aux[d * 3 + 0] + a1 * W_aux[d * 3 + 1]
                           + a2 * W_aux[d * 3 + 2];
        float gate = 1.0f / (1.0f + expf(-z)) * 0.15f + 1.0f;
        e[h] = entity[(size_t)node * EMB_D + d] * gate;
    }
    float ss = wave_sum32(e[0] * e[0] + e[1] * e[1]);
    float inv = 1.0f / fmaxf(sqrtf(ss), 1e-12f);
    int tl = (node >= ENTITY_OFFSET) ? 1 : 0;

    #pragma unroll
    for (int h = 0; h < 2; ++h) {
        int d = lane + 32 * h;
        float v = e[h] * inv + type_embed[tl * EMB_D + d];
        ego[(size_t)node * EMB_D + d] = v;
        all_e[(size_t)node * CAT_D + d] = v;     // all_embed[0] is un-normed ego
    }
}

// ---------------------------------------------------------------------------
// Kernel 2: COO SpMM scatter  side[row,:] += val * ego[col,:]
// One wave per edge; lane owns dims {lane, lane+32}.  ego table (25.6MB)
// lives in the 192MB L2, so the gather/atomic traffic stays on-chip.
// ---------------------------------------------------------------------------
__global__ __launch_bounds__(256)
void k_spmm(const int* __restrict__ rows, const int* __restrict__ cols,
            const float* __restrict__ vals, const float* __restrict__ ego,
            float* __restrict__ side, int E) {
    int gt = blockIdx.x * blockDim.x + threadIdx.x;
    int e = gt >> 5;
    if (e >= E) return;
    int lane = threadIdx.x & 31;

    // warm the L2 path for the next edge's gather row (global_prefetch_b8)
    int en = e + 8 < E ? e + 8 : E - 1;
    __builtin_prefetch(ego + (size_t)cols[en] * EMB_D, 0, 1);

    float v = vals[e];
    const float* src = ego + (size_t)cols[e] * EMB_D;
    float* dst = side + (size_t)rows[e] * EMB_D;
    atomicAdd(&dst[lane], v * src[lane]);
    atomicAdd(&dst[lane + 32], v * src[lane + 32]);
}

// ---------------------------------------------------------------------------
// Kernel 3: per-node transform via WMMA.
//   sum = leaky((ego+side) @ W1.T + b1);  bi = leaky((ego*side) @ W2.T + b2)
//   ego_out = sum + bi        (in-place: block reads its 16 nodes first)
// Block = 128 threads (4 waves) = one 16-node tile.  Raw ego/side tiles are
// staged into LDS with GLOBAL_LOAD_ASYNC_TO_LDS_B128; the e+s / e*s forms
// are built on the fly in the K loop (VALU co-executes with XDL WMMA).
// Wave w owns output columns [16w,16w+16).
// ---------------------------------------------------------------------------
template <int DIN, int DOUT>
__global__ __launch_bounds__(128)
void k_transform(const float* __restrict__ ego, const float* __restrict__ side,
                 const float* __restrict__ W1, const float* __restrict__ b1,
                 const float* __restrict__ W2, const float* __restrict__ b2,
                 float* __restrict__ ego_out, int N) {
    // row stride 68 floats: 16B-aligned rows (async b128 stores) AND
    // conflict-free column reads (bank = (4m + k) mod 64, distinct per m).
    constexpr int STR = DIN + 4;
    __shared__ float ego_s[16][STR];
    __shared__ float side_s[16][STR];

    int node0 = blockIdx.x * 16;
    // 16*DIN floats per array; one b128 (4 floats) per lane per iteration.
    constexpr int GROUPS = 16 * DIN / 4;
    for (int g = threadIdx.x; g < GROUPS; g += blockDim.x) {
        int t = g * 4;
        int m = t / DIN, d = t % DIN;        // d % 4 == 0, row-contained
        int node = node0 + m; if (node >= N) node = N - 1;
        async_copy_b128((unsigned)(uintptr_t)&ego_s[m][d],
                        ego + (size_t)node * EMB_D + d);
        async_copy_b128((unsigned)(uintptr_t)&side_s[m][d],
                        side + (size_t)node * EMB_D + d);
    }
    wait_asynccnt0();        // this wave's async writes have landed in LDS
    __syncthreads();         // ... and are visible to all waves

    const int wave = threadIdx.x >> 5;
    if (wave >= DOUT / 16) return;            // wave-uniform; no barriers after
    const int lane  = threadIdx.x & 31;
    const int mrow  = lane & 15;              // A: M = lane%16
    const int khalf = (lane >> 4) << 1;       // A/B: K = 2*(lane/16) + {0,1}
    const int n     = wave * 16 + (lane & 15);// B/C: N = lane%16

    v8f cs = {}; v8f cb = {};
    #pragma unroll
    for (int kk = 0; kk < DIN; kk += 4) {
        int k = kk + khalf;
        float e0 = ego_s[mrow][k],  e1 = ego_s[mrow][k + 1];
        float s0 = side_s[mrow][k], s1 = side_s[mrow][k + 1];
        v2f a1; a1.x = e0 + s0; a1.y = e1 + s1;
        v2f w1; w1.x = W1[n * DIN + k]; w1.y = W1[n * DIN + k + 1];
        cs = wmma_f32_16x16x4(a1, w1, cs);
        v2f a2; a2.x = e0 * s0; a2.y = e1 * s1;
        v2f w2; w2.x = W2[n * DIN + k]; w2.y = W2[n * DIN + k + 1];
        cb = wmma_f32_16x16x4(a2, w2, cb);
    }

    float bb1 = b1[n], bb2 = b2[n];
    #pragma unroll
    for (int v = 0; v < 8; ++v) {
        int m = v + ((lane >> 4) << 3);       // C/D: M = v + 8*(lane/16)
        int node = node0 + m;
        float r = leaky01(cs[v] + bb1) + leaky01(cb[v] + bb2);
        if (node < N) ego_out[(size_t)node * EMB_D + n] = r;
    }
}

// ---------------------------------------------------------------------------
// Kernel 4: l2-normalize ego rows (Dl dims, stride EMB_D) into all_e[:,off:].
// ---------------------------------------------------------------------------
__global__ __launch_bounds__(256)
void k_l2norm(const float* __restrict__ ego, float* __restrict__ all_e,
              int N, int Dl, int off) {
    int gt = blockIdx.x * blockDim.x + threadIdx.x;
    int node = gt >> 5;
    if (node >= N) return;
    int lane = threadIdx.x & 31;

    float v0 = (lane      < Dl) ? ego[(size_t)node * EMB_D + lane]      : 0.0f;
    float v1 = (lane + 32 < Dl) ? ego[(size_t)node * EMB_D + lane + 32] : 0.0f;
    float ss = wave_sum32(v0 * v0 + v1 * v1);
    float inv = 1.0f / fmaxf(sqrtf(ss), 1e-12f);
    if (lane < Dl)      all_e[(size_t)node * CAT_D + off + lane]      = v0 * inv;
    if (lane + 32 < Dl) all_e[(size_t)node * CAT_D + off + lane + 32] = v1 * inv;
}

// ---------------------------------------------------------------------------
// Kernel 5: score = all_e[user_ids] @ all_e[item_ids].T  via WMMA f32.
// Block = 128 threads (4 waves): one 16-user tile x four 16-item tiles.
// K = 160 -> 40 wmma steps.  all_e (64MB) fits L2, so the row gathers hit.
// ---------------------------------------------------------------------------
__global__ __launch_bounds__(128)
void k_score(const float* __restrict__ all_e, const int* __restrict__ uids,
             const int* __restrict__ iids, float* __restrict__ out,
             int NU, int NI) {
    const int wave = threadIdx.x >> 5;
    const int lane = threadIdx.x & 31;
    const int u0 = blockIdx.y * 16;
    const int n0 = (blockIdx.x * 4 + wave) * 16;
    if (n0 >= NI) return;                      // wave-uniform

    const int khalf = (lane >> 4) << 1;
    int urow = u0 + (lane & 15); if (urow >= NU) urow = NU - 1;
    int irow = n0 + (lane & 15); if (irow >= NI) irow = NI - 1;
    const float* ua = all_e + (size_t)uids[urow] * CAT_D;  // A: M=lane%16
    const float* ia = all_e + (size_t)iids[irow] * CAT_D;  // B: N=lane%16

    v8f c = {};
    #pragma unroll
    for (int kk = 0; kk < CAT_D; kk += 4) {
        int k = kk + khalf;
        v2f a; a.x = ua[k]; a.y = ua[k + 1];
        v2f b; b.x = ia[k]; b.y = ia[k + 1];
        c = wmma_f32_16x16x4(a, b, c);
    }

    #pragma unroll
    for (int v = 0; v < 8; ++v) {
        int m = u0 + v + ((lane >> 4) << 3);
        int n = n0 + (lane & 15);
        if (m < NU && n < NI) out[(size_t)m * NI + n] = c[v];
    }
}

// ---------------------------------------------------------------------------
extern "C" void kernel_launch(void* const* d_in, const int* in_sizes, int n_in,
                              void* d_out, int out_size, void* d_ws, size_t ws_size,
                              hipStream_t stream) {
    const int*   user_ids = (const int*)  d_in[0];
    const int*   item_ids = (const int*)  d_in[1];
    const float* aux      = (const float*)d_in[2];
    const int*   A_rows   = (const int*)  d_in[3];
    const int*   A_cols   = (const int*)  d_in[4];
    const float* A_vals   = (const float*)d_in[5];
    const float* entity   = (const float*)d_in[6];
    const float* type_e   = (const float*)d_in[7];
    const float* W_aux    = (const float*)d_in[8];
    const float* b_aux    = (const float*)d_in[9];
    const float* W1_0     = (const float*)d_in[10];
    const float* b1_0     = (const float*)d_in[11];
    const float* W2_0     = (const float*)d_in[12];
    const float* b2_0     = (const float*)d_in[13];
    const float* W1_1     = (const float*)d_in[14];
    const float* b1_1     = (const float*)d_in[15];
    const float* W2_1     = (const float*)d_in[16];
    const float* b2_1     = (const float*)d_in[17];

    const int N  = in_sizes[6] / EMB_D;   // 100000
    const int E  = in_sizes[3];           // 1600000
    const int NU = in_sizes[0];           // 1024
    const int NI = in_sizes[1];           // 4096

    float* all_e = (float*)d_ws;                         // [N,160]
    float* ego   = all_e + (size_t)N * CAT_D;            // [N,64]
    float* side  = ego   + (size_t)N * EMB_D;            // [N,64]

    const unsigned nodeBlocks = (unsigned)((N + 7) / 8);          // wave/node
    const unsigned edgeBlocks = (unsigned)(((size_t)E * 32 + 255) / 256);
    const unsigned tileBlocks = (unsigned)((N + 15) / 16);

    // fusion: gate, normalize, type embedding
    k_fuse<<<nodeBlocks, 256, 0, stream>>>(aux, entity, type_e, W_aux, b_aux,
                                           ego, all_e, N);

    // ---- layer 0 (64 -> 64) ----
    hipMemsetAsync(side, 0, (size_t)N * EMB_D * sizeof(float), stream);
    k_spmm<<<edgeBlocks, 256, 0, stream>>>(A_rows, A_cols, A_vals, ego, side, E);
    k_transform<64, 64><<<tileBlocks, 128, 0, stream>>>(ego, side, W1_0, b1_0,
                                                        W2_0, b2_0, ego, N);
    k_l2norm<<<nodeBlocks, 256, 0, stream>>>(ego, all_e, N, 64, 64);

    // ---- layer 1 (64 -> 32) ----
    hipMemsetAsync(side, 0, (size_t)N * EMB_D * sizeof(float), stream);
    k_spmm<<<edgeBlocks, 256, 0, stream>>>(A_rows, A_cols, A_vals, ego, side, E);
    k_transform<64, 32><<<tileBlocks, 128, 0, stream>>>(ego, side, W1_1, b1_1,
                                                        W2_1, b2_1, ego, N);
    k_l2norm<<<nodeBlocks, 256, 0, stream>>>(ego, all_e, N, 32, 128);

    // ---- scoring GEMM ----
    dim3 sg((NI + 63) / 64, (NU + 15) / 16);
    k_score<<<sg, 128, 0, stream>>>(all_e, user_ids, item_ids, (float*)d_out,
                                    NU, NI);
}